// MetaQDA_16982300688663
// MI455X (gfx1250) — compile-verified
//
#include <hip/hip_runtime.h>
#include <math.h>

#define D     1024
#define CLS   32
#define NSUP  1024
#define NQ    1024
#define REGP  0.5f

typedef __attribute__((ext_vector_type(2))) float v2f;
typedef __attribute__((ext_vector_type(8))) float v8f;

// ---------------- WMMA helpers (fp32 16x16x4) ----------------

__device__ __forceinline__ v8f wmma4(v2f a, v2f b, v8f c) {
  return __builtin_amdgcn_wmma_f32_16x16x4_f32(false, a, false, b, (short)0, c, false, false);
}

// C/D frag: VGPR r -> M = m0 + r (lanes 0-15) or m0 + 8 + r (lanes 16-31), N = n0 + (lane&15)
template <typename ST>
__device__ __forceinline__ void store_frag(v8f acc, int m0, int n0, ST st) {
  int lane = threadIdx.x & 31;
  int n  = n0 + (lane & 15);
  int mb = m0 + ((lane >> 4) << 3);
#pragma unroll
  for (int r = 0; r < 8; ++r) st(mb + r, n, acc[r]);
}

// One wave computes a 32x32 tile from LDS-staged panels (K=32):
// As[m][k] (rows = M), Bs[n][k] (rows = N, i.e. B pre-transposed).
__device__ __forceinline__ void mma32x32_lds(const float (*__restrict__ As)[36],
                                             const float (*__restrict__ Bs)[36],
                                             int am0, int bn0, v8f acc[2][2]) {
  int lane = threadIdx.x & 31;
  int half = lane >> 4;
  int l15  = lane & 15;
#pragma unroll
  for (int k0 = 0; k0 < 32; k0 += 4) {
    int kk = k0 + 2 * half;
    v2f a0 = *(const v2f*)&As[am0 + l15][kk];
    v2f a1 = *(const v2f*)&As[am0 + 16 + l15][kk];
    v2f b0 = *(const v2f*)&Bs[bn0 + l15][kk];
    v2f b1 = *(const v2f*)&Bs[bn0 + 16 + l15][kk];
    acc[0][0] = wmma4(a0, b0, acc[0][0]);
    acc[0][1] = wmma4(a0, b1, acc[0][1]);
    acc[1][0] = wmma4(a1, b0, acc[1][0]);
    acc[1][1] = wmma4(a1, b1, acc[1][1]);
  }
}

// ---------------- CDNA5 async global->LDS copy (VGPR-bypassing, ASYNCcnt) ----------------
// GVS mode: mem = SGPR64 + VGPR_I32 offset; LDS dest byte address in VGPR.
__device__ __forceinline__ void async_copy16(unsigned lds, unsigned goff, const float* base) {
  asm volatile("global_load_async_to_lds_b128 %0, %1, %2"
               :: "v"(lds), "v"(goff), "s"(base) : "memory");
}
__device__ __forceinline__ void wait_async0() {
  asm volatile("s_wait_asynccnt 0x0" ::: "memory");
}

// ---------------- scalar setup ----------------
// scal: [0]=kappa_, [1]=nu_, [2+c]=Nj, [2+C+c]=kN, [2+2C+c]=scale,
//       [2+3C+c]=common, [2+4C+c]=bias_base, [2+5C+c]=logdet accum
__global__ void k_scalars(const int* __restrict__ labels, const float* __restrict__ kappa,
                          const float* __restrict__ nu, float* __restrict__ scal,
                          int* __restrict__ idxl) {
  int c = threadIdx.x;
  float kap = fabsf(kappa[0]) + 1e-6f;
  float nu_ = fmaxf(nu[0], (float)D - 1.0f + 1e-6f);
  int cnt = 0;
  for (int n = 0; n < NSUP; ++n)
    if (labels[n] == c) idxl[c * NSUP + cnt++] = n;
  float Nj = (float)cnt;
  float kN = kap + Nj;
  float common = nu_ + Nj + 1.0f - (float)D;
  float scale = (kN + 1.0f) / ((nu_ + Nj - (float)D + 1.0f) * kN);
  float biasb = lgammaf(0.5f * (common + (float)D)) - lgammaf(0.5f * common)
              - 0.5f * (float)D * logf(common);
  if (c == 0) { scal[0] = kap; scal[1] = nu_; }
  scal[2 + 0 * CLS + c] = Nj;
  scal[2 + 1 * CLS + c] = kN;
  scal[2 + 2 * CLS + c] = scale;
  scal[2 + 3 * CLS + c] = common;
  scal[2 + 4 * CLS + c] = biasb;
  scal[2 + 5 * CLS + c] = 0.0f;
  idxl[CLS * NSUP + c] = cnt;
}

__global__ void k_mu(const float* __restrict__ sx, const float* __restrict__ m,
                     const float* __restrict__ scal, const int* __restrict__ idxl,
                     float* __restrict__ mu) {
  int d = blockIdx.x * 256 + threadIdx.x;
  int c = blockIdx.y;
  int cnt = idxl[CLS * NSUP + c];
  float kap = scal[0];
  float kN  = scal[2 + CLS + c];
  float s = 0.f;
  for (int i = 0; i < cnt; ++i) s += sx[(size_t)idxl[c * NSUP + i] * D + d];
  mu[c * D + d] = (kap * m[d] + s) / kN;
}

// ---------------- prior_S = L * L^T (64x64 tiles, LDS-staged, branchless) ----------------
__global__ void k_priorS(const float* __restrict__ dg, const float* __restrict__ lw,
                         float* __restrict__ S) {
  __shared__ float As[64][36];
  __shared__ float Bs[64][36];
  int tM = blockIdx.x * 64, tN = blockIdx.y * 64;
  int t = threadIdx.x;
  int w = t >> 5;
  int wM = (w >> 1) * 32, wN = (w & 1) * 32;
  v8f acc[2][2] = {};
  int tmin = tM < tN ? tM : tN;
  int kmax = tmin + 64; if (kmax > D) kmax = D;
  for (int kc = 0; kc < kmax; kc += 32) {
#pragma unroll
    for (int v = 0; v < 4; ++v) {
      int e   = t + v * 128;
      int row = e >> 3;
      int kq  = (e & 7) * 4;
      int k0  = kc + kq;
      {
        int gr = tM + row;
        float4 lv = *(const float4*)&lw[(size_t)gr * D + k0];
        float  dv = fabsf(dg[gr]);
        float4 o;
        o.x = (k0 + 0 < gr) ? lv.x : ((k0 + 0 == gr) ? dv : 0.f);
        o.y = (k0 + 1 < gr) ? lv.y : ((k0 + 1 == gr) ? dv : 0.f);
        o.z = (k0 + 2 < gr) ? lv.z : ((k0 + 2 == gr) ? dv : 0.f);
        o.w = (k0 + 3 < gr) ? lv.w : ((k0 + 3 == gr) ? dv : 0.f);
        *(float4*)&As[row][kq] = o;
      }
      {
        int gr = tN + row;
        float4 lv = *(const float4*)&lw[(size_t)gr * D + k0];
        float  dv = fabsf(dg[gr]);
        float4 o;
        o.x = (k0 + 0 < gr) ? lv.x : ((k0 + 0 == gr) ? dv : 0.f);
        o.y = (k0 + 1 < gr) ? lv.y : ((k0 + 1 == gr) ? dv : 0.f);
        o.z = (k0 + 2 < gr) ? lv.z : ((k0 + 2 == gr) ? dv : 0.f);
        o.w = (k0 + 3 < gr) ? lv.w : ((k0 + 3 == gr) ? dv : 0.f);
        *(float4*)&Bs[row][kq] = o;
      }
    }
    __syncthreads();
    mma32x32_lds(As, Bs, wM, wN, acc);
    __syncthreads();
  }
#pragma unroll
  for (int i = 0; i < 2; ++i)
#pragma unroll
    for (int j = 0; j < 2; ++j)
      store_frag(acc[i][j], tM + wM + 16 * i, tN + wN + 16 * j,
                 [&](int r, int cc, float v) { S[(size_t)r * D + cc] = v; });
}

// ---------------- sigma_c = scale*(prior + Xc^T Xc + kap*m m^T - kN*mu mu^T) ----------------
__global__ void k_sigma(const float* __restrict__ sx, const float* __restrict__ m,
                        const float* __restrict__ mu, const float* __restrict__ scal,
                        const int* __restrict__ idxl, const float* __restrict__ prior,
                        float* __restrict__ A, int c) {
  __shared__ float As[64][36];
  __shared__ float Bs[64][36];
  int tM = blockIdx.x * 64, tN = blockIdx.y * 64;
  int t = threadIdx.x;
  int w = t >> 5;
  int wM = (w >> 1) * 32, wN = (w & 1) * 32;
  const int* idx = idxl + c * NSUP;
  int cnt = idxl[CLS * NSUP + c];
  v8f acc[2][2] = {};
  for (int kc = 0; kc < cnt; kc += 32) {
#pragma unroll
    for (int v = 0; v < 4; ++v) {
      int e  = t + v * 128;
      int kk = e >> 4;
      int dq = (e & 15) * 4;
      int k  = kc + kk;
      bool ok = (k < cnt);
      int  n  = idx[ok ? k : 0];
      float msk = ok ? 1.f : 0.f;
      float4 xa = *(const float4*)&sx[(size_t)n * D + tM + dq];
      float4 xb = *(const float4*)&sx[(size_t)n * D + tN + dq];
      As[dq + 0][kk] = xa.x * msk; As[dq + 1][kk] = xa.y * msk;
      As[dq + 2][kk] = xa.z * msk; As[dq + 3][kk] = xa.w * msk;
      Bs[dq + 0][kk] = xb.x * msk; Bs[dq + 1][kk] = xb.y * msk;
      Bs[dq + 2][kk] = xb.z * msk; Bs[dq + 3][kk] = xb.w * msk;
    }
    __syncthreads();
    mma32x32_lds(As, Bs, wM, wN, acc);
    __syncthreads();
  }
  float kap = scal[0];
  float kN  = scal[2 + CLS + c];
  float scl = scal[2 + 2 * CLS + c];
  const float* muc = mu + c * D;
#pragma unroll
  for (int i = 0; i < 2; ++i)
#pragma unroll
    for (int j = 0; j < 2; ++j)
      store_frag(acc[i][j], tM + wM + 16 * i, tN + wN + 16 * j, [&](int r, int cc, float v) {
        float s = prior[(size_t)r * D + cc] + v + kap * m[r] * m[cc] - kN * muc[r] * muc[cc];
        A[(size_t)r * D + cc] = s * scl;
      });
}

// ---------------- Cholesky 32x32 diagonal factorization (1 wave) ----------------
__global__ void k_potf2(float* __restrict__ A, float* __restrict__ scal, int c, int kB) {
  __shared__ float T[32][33];
  __shared__ float lred[32];
  int t = threadIdx.x;
  for (int j = 0; j < 32; ++j) T[t][j] = A[(size_t)(kB + t) * D + kB + j];
  float myld = 0.f;
  for (int j = 0; j < 32; ++j) {
    __syncthreads();
    if (t == j) { float dd = sqrtf(T[j][j]); T[j][j] = dd; myld = logf(dd); }
    __syncthreads();
    float dd = T[j][j];
    if (t > j) T[t][j] = T[t][j] / dd;
    __syncthreads();
    if (t > j) {
      float l = T[t][j];
      for (int p = j + 1; p <= t; ++p) T[t][p] -= l * T[p][j];
    }
  }
  __syncthreads();
  lred[t] = myld;
  __syncthreads();
  for (int j = 0; j < 32; ++j) A[(size_t)(kB + t) * D + kB + j] = T[t][j];
  if (t == 0) {
    float s = 0.f;
    for (int j = 0; j < 32; ++j) s += lred[j];
    scal[2 + 5 * CLS + c] += 2.f * s;
  }
}

// panel: L21 = A21 * G_kk^-T (per-row forward substitution)
__global__ void k_trsm(float* __restrict__ A, int kB) {
  __shared__ float G[32][33];
  int t = threadIdx.x;
  for (int i = t; i < 1024; i += 256)
    G[i >> 5][i & 31] = A[(size_t)(kB + (i >> 5)) * D + kB + (i & 31)];
  __syncthreads();
  int r = kB + 32 + blockIdx.x * 256 + t;
  if (r >= D) return;
  float x[32];
#pragma unroll
  for (int j = 0; j < 32; ++j) x[j] = A[(size_t)r * D + kB + j];
#pragma unroll
  for (int j = 0; j < 32; ++j) {
    float s = x[j];
    for (int p = 0; p < j; ++p) s -= x[p] * G[j][p];
    x[j] = s / G[j][j];
  }
#pragma unroll
  for (int j = 0; j < 32; ++j) A[(size_t)r * D + kB + j] = x[j];
}

// trailing update, lower tiles only: A22 -= L21 * L21^T (K=32, LDS-staged WMMA)
__global__ void k_trailing(float* __restrict__ A, int kB) {
  __shared__ float As[32][36];
  __shared__ float Bs[32][36];
  int base = kB + 32;
  int bid = blockIdx.x;
  int i = (int)((sqrtf(8.f * (float)bid + 1.f) - 1.f) * 0.5f);
  while ((i + 1) * (i + 2) / 2 <= bid) ++i;
  while (i * (i + 1) / 2 > bid) --i;
  int j = bid - i * (i + 1) / 2;
  int rM = base + i * 32, rN = base + j * 32;
  int t = threadIdx.x;
#pragma unroll
  for (int v = 0; v < 8; ++v) {
    int e   = t + v * 32;
    int row = e >> 3;
    int kq  = (e & 7) * 4;
    *(float4*)&As[row][kq] = *(const float4*)&A[(size_t)(rM + row) * D + kB + kq];
    *(float4*)&Bs[row][kq] = *(const float4*)&A[(size_t)(rN + row) * D + kB + kq];
  }
  __syncthreads();
  v8f acc[2][2] = {};
  mma32x32_lds(As, Bs, 0, 0, acc);
#pragma unroll
  for (int ii = 0; ii < 2; ++ii)
#pragma unroll
    for (int jj = 0; jj < 2; ++jj)
      store_frag(acc[ii][jj], rM + 16 * ii, rN + 16 * jj,
                 [&](int r, int cc, float v) { A[(size_t)r * D + cc] -= v; });
}

// blocked forward substitution, GEMM step (64 q-rows x 32 cols, 2 waves):
// U[:, jB..jB+32) = diff[:, jB..) - Y[:, 0..jB) * G[jB.., 0..jB)^T
// Double-buffered async global->LDS staging (ASYNCcnt) overlapped with WMMA.
__global__ void k_ygemm(const float* __restrict__ A, float* __restrict__ Y,
                        const float* __restrict__ qx, const float* __restrict__ mu,
                        int c, int jB) {
  __shared__ float As[2][64][36];   // Y panel  [q][p]
  __shared__ float Bs[2][32][36];   // G panel  [n][p]
  int q0 = blockIdx.x * 64;
  int t = threadIdx.x;
  int w = t >> 5;
  v8f acc[2][2] = {};

  unsigned asBase = (unsigned)(unsigned long long)&As[0][0][0];
  unsigned bsBase = (unsigned)(unsigned long long)&Bs[0][0][0];
  const unsigned asStride = 64u * 36u * 4u;
  const unsigned bsStride = 32u * 36u * 4u;

  auto stage = [&](int buf, int kc) {
#pragma unroll
    for (int v = 0; v < 8; ++v) {       // 512 b128 units for As
      int e = t + v * 64; int row = e >> 3; int pq = (e & 7) * 4;
      unsigned lds = asBase + (unsigned)buf * asStride + (unsigned)(row * 36 + pq) * 4u;
      unsigned go  = (unsigned)(((q0 + row) * D + kc + pq) * 4);
      async_copy16(lds, go, Y);
    }
#pragma unroll
    for (int v = 0; v < 4; ++v) {       // 256 b128 units for Bs
      int e = t + v * 64; int row = e >> 3; int pq = (e & 7) * 4;
      unsigned lds = bsBase + (unsigned)buf * bsStride + (unsigned)(row * 36 + pq) * 4u;
      unsigned go  = (unsigned)(((jB + row) * D + kc + pq) * 4);
      async_copy16(lds, go, A);
    }
  };

  int nChunks = jB / 32;
  if (nChunks > 0) {
    stage(0, 0);
    int cur = 0;
    for (int ch = 0; ch < nChunks; ++ch) {
      wait_async0();
      __syncthreads();                                 // panel `cur` visible to all waves
      if (ch + 1 < nChunks) stage(cur ^ 1, (ch + 1) * 32);  // prefetch next while computing
      mma32x32_lds(As[cur], Bs[cur], w * 32, 0, acc);
      __syncthreads();                                 // all waves done with `cur`
      cur ^= 1;
    }
  }
  const float* muc = mu + c * D;
#pragma unroll
  for (int i = 0; i < 2; ++i)
#pragma unroll
    for (int j = 0; j < 2; ++j)
      store_frag(acc[i][j], q0 + w * 32 + 16 * i, 16 * j, [&](int q, int n, float v) {
        int dcol = jB + n;
        Y[(size_t)q * D + dcol] = (qx[(size_t)q * D + dcol] - muc[dcol]) - v;
      });
}

// triangular solve with 32x32 diagonal block, per-query row
__global__ void k_ytrsm(const float* __restrict__ A, float* __restrict__ Y, int jB) {
  __shared__ float G[32][33];
  int t = threadIdx.x;
  for (int i = t; i < 1024; i += 256)
    G[i >> 5][i & 31] = A[(size_t)(jB + (i >> 5)) * D + jB + (i & 31)];
  __syncthreads();
  int q = blockIdx.x * 256 + t;
  float x[32];
#pragma unroll
  for (int j = 0; j < 32; ++j) x[j] = Y[(size_t)q * D + jB + j];
#pragma unroll
  for (int j = 0; j < 32; ++j) {
    float s = x[j];
    for (int p = 0; p < j; ++p) s -= x[p] * G[j][p];
    x[j] = s / G[j][j];
  }
#pragma unroll
  for (int j = 0; j < 32; ++j) Y[(size_t)q * D + jB + j] = x[j];
}

// final: dist = (1-REG)*||y||^2 + REG*||diff||^2 ; preds = bias - 0.5*(common+D)*log1p(dist/common)
__global__ void k_dist(const float* __restrict__ Y, const float* __restrict__ qx,
                       const float* __restrict__ mu, const float* __restrict__ scal,
                       int c, float* __restrict__ out) {
  int q = blockIdx.x * 256 + threadIdx.x;
  const float* muc = mu + c * D;
  float sy = 0.f, sd = 0.f;
  for (int d = 0; d < D; d += 4) {
    float4 y  = *(const float4*)&Y[(size_t)q * D + d];
    float4 xq = *(const float4*)&qx[(size_t)q * D + d];
    float4 mm = *(const float4*)&muc[d];
    sy += y.x * y.x + y.y * y.y + y.z * y.z + y.w * y.w;
    float a = xq.x - mm.x, b = xq.y - mm.y, e = xq.z - mm.z, f = xq.w - mm.w;
    sd += a * a + b * b + e * e + f * f;
  }
  float dist   = (1.0f - REGP) * sy + REGP * sd;
  float common = scal[2 + 3 * CLS + c];
  float bias   = scal[2 + 4 * CLS + c] - 0.5f * scal[2 + 5 * CLS + c];
  out[(size_t)q * CLS + c] = bias - 0.5f * (common + (float)D) * log1pf(dist / common);
}

// ---------------- host orchestration ----------------
extern "C" void kernel_launch(void* const* d_in, const int* in_sizes, int n_in,
                              void* d_out, int out_size, void* d_ws, size_t ws_size,
                              hipStream_t stream) {
  const float* sx    = (const float*)d_in[0];
  const float* qx    = (const float*)d_in[1];
  const int*   lb    = (const int*)d_in[2];
  const float* m     = (const float*)d_in[3];
  const float* kappa = (const float*)d_in[4];
  const float* nu    = (const float*)d_in[5];
  const float* tdiag = (const float*)d_in[6];
  const float* tlow  = (const float*)d_in[7];
  float* out = (float*)d_out;

  char* ws = (char*)d_ws;
  size_t off = 0;
  auto alloc = [&](size_t bytes) -> void* {
    void* p = ws + off;
    off += (bytes + 255) & ~(size_t)255;
    return p;
  };
  float* prior = (float*)alloc((size_t)D * D * 4);
  float* mu    = (float*)alloc((size_t)CLS * D * 4);
  float* scal  = (float*)alloc((2 + 6 * CLS) * 4);
  int*   idxl  = (int*)alloc(((size_t)CLS * NSUP + CLS) * 4);
  float* A     = (float*)alloc((size_t)D * D * 4);
  float* Y     = (float*)alloc((size_t)NQ * D * 4);
  (void)in_sizes; (void)n_in; (void)out_size; (void)ws_size;

  k_scalars<<<1, CLS, 0, stream>>>(lb, kappa, nu, scal, idxl);
  k_mu<<<dim3(D / 256, CLS), 256, 0, stream>>>(sx, m, scal, idxl, mu);
  k_priorS<<<dim3(D / 64, D / 64), 128, 0, stream>>>(tdiag, tlow, prior);

  for (int c = 0; c < CLS; ++c) {
    k_sigma<<<dim3(D / 64, D / 64), 128, 0, stream>>>(sx, m, mu, scal, idxl, prior, A, c);
    for (int k = 0; k < D / 32; ++k) {
      int kB = k * 32;
      k_potf2<<<1, 32, 0, stream>>>(A, scal, c, kB);
      int R = D - kB - 32;
      if (R > 0) {
        k_trsm<<<(R + 255) / 256, 256, 0, stream>>>(A, kB);
        int nt = R / 32;
        k_trailing<<<nt * (nt + 1) / 2, 32, 0, stream>>>(A, kB);
      }
    }
    for (int j = 0; j < D / 32; ++j) {
      int jB = j * 32;
      k_ygemm<<<NQ / 64, 64, 0, stream>>>(A, Y, qx, mu, c, jB);
      k_ytrsm<<<NQ / 256, 256, 0, stream>>>(A, Y, jB);
    }
    k_dist<<<NQ / 256, 256, 0, stream>>>(Y, qx, mu, scal, c, out);
  }
}